// Encoder_14955076125205
// MI455X (gfx1250) — compile-verified
//
#include <hip/hip_runtime.h>
#include <hip/hip_bf16.h>

#define N_NODES 50000
#define N_EDGES 800000
#define IN_DIM  512
#define HID_DIM 128
#define M_TILES (N_NODES / 16)   // 3125, exact

// K-pair-interleaved LDS layout for W chunk: pair stride padded to 288 floats
// (288 mod 64 == 32) so half-wave 0 (banks 0-31) and half-wave 1 (banks 32-63)
// hit disjoint LDS banks on ds_load_b64.
#define PAIR_STRIDE 288
#define N_PAIRS     16           // 32 K-values per chunk = 16 pairs

typedef float v2f __attribute__((ext_vector_type(2)));
typedef float v8f __attribute__((ext_vector_type(8)));

// ---------------- init: zero output accumulator, deg = 1.0 (self-loop) -------
__global__ void init_kernel(float* __restrict__ out, float* __restrict__ deg) {
    int i = blockIdx.x * blockDim.x + threadIdx.x;
    if (i < N_NODES * HID_DIM) out[i] = 0.0f;
    if (i < N_NODES) deg[i] = 1.0f;
}

// ---------------- degree accumulation ----------------------------------------
__global__ void deg_kernel(const int* __restrict__ dst, float* __restrict__ deg) {
    int e = blockIdx.x * blockDim.x + threadIdx.x;
    if (e < N_EDGES) atomicAdd(&deg[dst[e]], 1.0f);
}

// ---------------- dinv = deg^-1/2 --------------------------------------------
__global__ void dinv_kernel(const float* __restrict__ deg, float* __restrict__ dinv) {
    int i = blockIdx.x * blockDim.x + threadIdx.x;
    if (i < N_NODES) dinv[i] = rsqrtf(deg[i]);
}

// ---------------- h = x @ W via V_WMMA_F32_16X16X4_F32 -----------------------
// 256 threads = 8 waves; wave w computes M-tile (blockIdx.x*8 + w), all 8 N-tiles.
// W K-chunk (32 x 128) staged in LDS in K-pair-interleaved layout so each
// B fragment is one aligned ds_load_b64.
__global__ __launch_bounds__(256) void gemm_kernel(const float* __restrict__ x,
                                                   const float* __restrict__ W,
                                                   float* __restrict__ h) {
    __shared__ float ldsW[N_PAIRS * PAIR_STRIDE];   // 18.4 KB

    const int lane   = threadIdx.x & 31;
    const int waveId = threadIdx.x >> 5;
    const int mTile  = blockIdx.x * 8 + waveId;
    const bool valid = (mTile < M_TILES);

    const int half = lane >> 4;        // 0: K pair {k,k+1}, 1: K pair {k+2,k+3}
    const int l16  = lane & 15;        // M (for A) / N (for B) within tile

    // staging-load coordinates
    const int c  = threadIdx.x & 127;  // W column 0..127
    const int ph = threadIdx.x >> 7;   // pair-group 0 or 1

    v8f acc[8];
#pragma unroll
    for (int t = 0; t < 8; ++t) acc[t] = (v8f){0.f,0.f,0.f,0.f,0.f,0.f,0.f,0.f};

    const int row = mTile * 16 + l16;           // A-matrix row for this lane
    const float* xrow = x + (size_t)row * IN_DIM;

    for (int kc = 0; kc < IN_DIM; kc += 32) {
        // Stage W[kc..kc+31][0..127] interleaved by K-pair:
        // lds[pair*PAIR_STRIDE + n*2 + 0/1] = W[kc+2*pair + 0/1][n]
        // Each global load is coalesced (consecutive threads sweep n).
#pragma unroll
        for (int pp = 0; pp < 8; ++pp) {
            const int pair = ph * 8 + pp;
            float w0 = W[(size_t)(kc + 2 * pair    ) * HID_DIM + c];
            float w1 = W[(size_t)(kc + 2 * pair + 1) * HID_DIM + c];
            *(v2f*)(ldsW + pair * PAIR_STRIDE + c * 2) = (v2f){w0, w1};
        }
        __syncthreads();

        if (valid) {
            if (kc + 32 < IN_DIM)
                __builtin_prefetch(xrow + kc + 32, 0, 3);   // global_prefetch_b8
#pragma unroll
            for (int k = 0; k < 32; k += 4) {
                const int kk   = k + 2 * half;              // this lane's K pair base
                const int pair = (k >> 1) + half;           // LDS pair index
                // A fragment: lane holds x[row][kc+kk], x[row][kc+kk+1]
                v2f a = *(const v2f*)(xrow + kc + kk);
                const float* bbase = ldsW + (size_t)pair * PAIR_STRIDE;
#pragma unroll
                for (int t = 0; t < 8; ++t) {
                    // B fragment: contiguous aligned pair -> single ds_load_b64
                    v2f b = *(const v2f*)(bbase + (t * 16 + l16) * 2);
                    acc[t] = __builtin_amdgcn_wmma_f32_16x16x4_f32(
                        /*neg_a=*/false, a, /*neg_b=*/false, b,
                        /*c_mod=*/(short)0, acc[t],
                        /*reuse_a=*/false, /*reuse_b=*/false);
                }
            }
        }
        __syncthreads();
    }

    if (valid) {
        // C/D layout: VGPR r, lanes 0-15 -> M=r, lanes 16-31 -> M=r+8; N = l16
#pragma unroll
        for (int t = 0; t < 8; ++t) {
#pragma unroll
            for (int r = 0; r < 8; ++r) {
                int m = mTile * 16 + r + 8 * half;
                h[(size_t)m * HID_DIM + t * 16 + l16] = acc[t][r];
            }
        }
    }
}

// ---------------- edge aggregation: one wave per edge ------------------------
__global__ __launch_bounds__(256) void aggregate_kernel(const float* __restrict__ h,
                                                        const int* __restrict__ src,
                                                        const int* __restrict__ dst,
                                                        const float* __restrict__ dinv,
                                                        float* __restrict__ out) {
    int tid  = blockIdx.x * blockDim.x + threadIdx.x;
    int e    = tid >> 5;
    int lane = tid & 31;
    if (e >= N_EDGES) return;
    int s = src[e];
    int d = dst[e];
    float norm = dinv[s] * dinv[d];
    float4 v = ((const float4*)(h + (size_t)s * HID_DIM))[lane];  // 32 lanes x 16B = 512B row
    float* o = out + (size_t)d * HID_DIM + lane * 4;
    atomicAdd(o + 0, v.x * norm);
    atomicAdd(o + 1, v.y * norm);
    atomicAdd(o + 2, v.z * norm);
    atomicAdd(o + 3, v.w * norm);
}

// ---------------- self-loop + bias + PReLU -----------------------------------
__global__ void finalize_kernel(const float* __restrict__ h,
                                const float* __restrict__ dinv,
                                const float* __restrict__ b,
                                const float* __restrict__ prelu_w,
                                float* __restrict__ out) {
    int i = blockIdx.x * blockDim.x + threadIdx.x;
    if (i >= N_NODES * HID_DIM) return;
    int rowI = i / HID_DIM;
    int c    = i - rowI * HID_DIM;
    float di = dinv[rowI];
    float v  = out[i] + h[i] * di * di + b[c];
    out[i]   = (v > 0.0f) ? v : prelu_w[c] * v;
}

extern "C" void kernel_launch(void* const* d_in, const int* in_sizes, int n_in,
                              void* d_out, int out_size, void* d_ws, size_t ws_size,
                              hipStream_t stream) {
    const float* x       = (const float*)d_in[0];
    const int*   eidx    = (const int*)d_in[1];
    const float* W       = (const float*)d_in[2];
    const float* b       = (const float*)d_in[3];
    const float* prelu_w = (const float*)d_in[4];
    float*       out     = (float*)d_out;

    const int* src = eidx;            // edge_index[0]
    const int* dst = eidx + N_EDGES;  // edge_index[1]

    // workspace layout: h | deg | dinv
    char* ws    = (char*)d_ws;
    float* h    = (float*)(ws);                                        // 25.6 MB
    float* deg  = (float*)(ws + (size_t)N_NODES * HID_DIM * 4);        // 200 KB
    float* dinv = (float*)(ws + (size_t)N_NODES * HID_DIM * 4 + (size_t)N_NODES * 4);

    const int TPB = 256;

    // 1) zero out accumulator, deg = 1 (self loop)
    init_kernel<<<(N_NODES * HID_DIM + TPB - 1) / TPB, TPB, 0, stream>>>(out, deg);

    // 2) degree accumulation over edges
    deg_kernel<<<(N_EDGES + TPB - 1) / TPB, TPB, 0, stream>>>(dst, deg);

    // 3) dinv = rsqrt(deg)
    dinv_kernel<<<(N_NODES + TPB - 1) / TPB, TPB, 0, stream>>>(deg, dinv);

    // 4) h = x @ W  (WMMA fp32); 8 M-tiles per block
    gemm_kernel<<<(M_TILES + 7) / 8, TPB, 0, stream>>>(x, W, h);

    // 5) gather-scale-scatter: one wave per edge
    aggregate_kernel<<<(N_EDGES * 32 + TPB - 1) / TPB, TPB, 0, stream>>>(h, src, dst, dinv, out);

    // 6) self-loop + bias + PReLU
    finalize_kernel<<<(N_NODES * HID_DIM + TPB - 1) / TPB, TPB, 0, stream>>>(h, dinv, b, prelu_w, out);
}